// BaseDepthVolumeModel_80736795230428
// MI455X (gfx1250) — compile-verified
//
#include <hip/hip_runtime.h>
#include <hip/hip_bf16.h>

// Problem constants (fixed by the reference)
#define NB 2
#define NV 4
#define HH 240
#define WW 320
#define DN 48

typedef __attribute__((ext_vector_type(2))) float v2f;
typedef __attribute__((ext_vector_type(8))) float v8f;

// ---------------------------------------------------------------------------
// Setup: per (n,v) fold all matrices into A = Ks*M33*Kd^-1 (3x3), b = Ks*t.
// ws layout per nv (16 floats): A[0..8] row-major, b[9..11],
// xs_src[12], ys_src[13], xs_dst[14], ys_dst[15].
// ---------------------------------------------------------------------------
__global__ void bdvm_setup_kernel(const float* __restrict__ ys_dst,
                                  const float* __restrict__ xs_dst,
                                  const float* __restrict__ ys_src,
                                  const float* __restrict__ xs_src,
                                  const float* __restrict__ dstK,
                                  const float* __restrict__ dstE,
                                  const float* __restrict__ srcK,
                                  const float* __restrict__ srcE,
                                  float* __restrict__ ws) {
  int nv = threadIdx.x;
  if (nv >= NB * NV) return;
  const float* Kd = dstK + nv * 9;
  const float* Ed = dstE + nv * 16;
  const float* Ks = srcK + nv * 9;
  const float* Es = srcE + nv * 16;

  // General 3x3 inverse of dst intrinsics (adjugate / det)
  float a00 = Kd[0], a01 = Kd[1], a02 = Kd[2];
  float a10 = Kd[3], a11 = Kd[4], a12 = Kd[5];
  float a20 = Kd[6], a21 = Kd[7], a22 = Kd[8];
  float det = a00 * (a11 * a22 - a12 * a21)
            - a01 * (a10 * a22 - a12 * a20)
            + a02 * (a10 * a21 - a11 * a20);
  float id = 1.0f / det;
  float iK[9];
  iK[0] = (a11 * a22 - a12 * a21) * id; iK[1] = (a02 * a21 - a01 * a22) * id; iK[2] = (a01 * a12 - a02 * a11) * id;
  iK[3] = (a12 * a20 - a10 * a22) * id; iK[4] = (a00 * a22 - a02 * a20) * id; iK[5] = (a02 * a10 - a00 * a12) * id;
  iK[6] = (a10 * a21 - a11 * a20) * id; iK[7] = (a01 * a20 - a00 * a21) * id; iK[8] = (a00 * a11 - a01 * a10) * id;

  // Rigid inverse of dst extrinsics: [R t; 0 1]^-1 = [R^T, -R^T t]
  float Rt[9], it3[3];
#pragma unroll
  for (int i = 0; i < 3; ++i)
#pragma unroll
    for (int j = 0; j < 3; ++j) Rt[i * 3 + j] = Ed[j * 4 + i];
#pragma unroll
  for (int i = 0; i < 3; ++i)
    it3[i] = -(Rt[i * 3 + 0] * Ed[0 * 4 + 3] + Rt[i * 3 + 1] * Ed[1 * 4 + 3] + Rt[i * 3 + 2] * Ed[2 * 4 + 3]);

  // M = src_E * inv(dst_E): take top 3 rows
  float M33[9], Mt[3];
#pragma unroll
  for (int i = 0; i < 3; ++i) {
#pragma unroll
    for (int j = 0; j < 3; ++j)
      M33[i * 3 + j] = Es[i * 4 + 0] * Rt[0 * 3 + j] + Es[i * 4 + 1] * Rt[1 * 3 + j] + Es[i * 4 + 2] * Rt[2 * 3 + j];
    Mt[i] = Es[i * 4 + 0] * it3[0] + Es[i * 4 + 1] * it3[1] + Es[i * 4 + 2] * it3[2] + Es[i * 4 + 3];
  }

  // A = Ks * M33 * iK ; b = Ks * Mt
  float T[9], A[9], b3[3];
#pragma unroll
  for (int i = 0; i < 3; ++i)
#pragma unroll
    for (int j = 0; j < 3; ++j)
      T[i * 3 + j] = Ks[i * 3 + 0] * M33[0 * 3 + j] + Ks[i * 3 + 1] * M33[1 * 3 + j] + Ks[i * 3 + 2] * M33[2 * 3 + j];
#pragma unroll
  for (int i = 0; i < 3; ++i)
#pragma unroll
    for (int j = 0; j < 3; ++j)
      A[i * 3 + j] = T[i * 3 + 0] * iK[0 * 3 + j] + T[i * 3 + 1] * iK[1 * 3 + j] + T[i * 3 + 2] * iK[2 * 3 + j];
#pragma unroll
  for (int i = 0; i < 3; ++i)
    b3[i] = Ks[i * 3 + 0] * Mt[0] + Ks[i * 3 + 1] * Mt[1] + Ks[i * 3 + 2] * Mt[2];

  float* o = ws + nv * 16;
#pragma unroll
  for (int i = 0; i < 9; ++i) o[i] = A[i];
  o[9] = b3[0]; o[10] = b3[1]; o[11] = b3[2];
  o[12] = xs_src[nv]; o[13] = ys_src[nv];
  o[14] = xs_dst[nv]; o[15] = ys_dst[nv];
}

// ---------------------------------------------------------------------------
// Main sweep. One wave32 per 16-pixel strip of one image row of one (n,v):
//   - one v_wmma_f32_16x16x4_f32 computes r = A @ p for all 16 pixels,
//     duplicated into rows 0-2 and 8-10 so every lane holds (rx,ry,rz) of
//     pixel lane%16 in d[0..2] with no cross-lane traffic.
//   - lanes 0-15 iterate depths 0..23; lanes 16-31 iterate depths 24..47.
//   - depth loop fully unrolled: it*dstep folds to literals (no int->float
//     converts, no loop-counter SALU); store offsets fold to constants.
//   - coalesced non-temporal stores (output streams past the 192MB L2).
// ---------------------------------------------------------------------------
__global__ __launch_bounds__(256) void bdvm_sweep_kernel(const float* __restrict__ ws,
                                                         float* __restrict__ out) {
  const int lane = threadIdx.x & 31;
  const int wave = threadIdx.x >> 5;
  const int strip = blockIdx.x * 8 + wave;         // NB*NV*HH*(WW/16) strips
  const int w16 = strip % (WW / 16);
  const int h = (strip / (WW / 16)) % HH;
  const int nv = strip / ((WW / 16) * HH);

  const float* P = ws + nv * 16;
  const int m = lane & 15;
  const bool lo = lane < 16;

  // A-operand (16x4 f32): lane holds row m=lane%16, VGPR0 = col (lo?0:2),
  // VGPR1 = col (lo?1:3). Rows 0-2 and 8-10 carry the 3x3; col 3 is zero.
  int row = (m < 3) ? m : ((m >= 8 && m < 11) ? (m - 8) : -1);
  v2f aop;
  aop.x = (row >= 0) ? P[row * 3 + (lo ? 0 : 2)] : 0.0f;
  aop.y = (row >= 0 && lo) ? P[row * 3 + 1] : 0.0f;

  // B-operand (4x16 f32): VGPR0 = {K0 row: x coords | K2 row: ones},
  //                       VGPR1 = {K1 row: y coords | K3 row: zeros}.
  const float xoff = P[14], yoff = P[15];
  v2f bop;
  bop.x = lo ? ((float)(w16 * 16 + m) + xoff) : 1.0f;
  bop.y = lo ? ((float)h + yoff) : 0.0f;

  v8f c = {};
  v8f d = __builtin_amdgcn_wmma_f32_16x16x4_f32(false, aop, false, bop,
                                                (short)0, c, false, false);
  const float rx = d[0], ry = d[1], rz = d[2];   // ray of pixel lane%16

  const float bx = P[9], by = P[10], bz = P[11];
  const float xs = P[12], ys = P[13];

  const float dstep = 9.5f / 47.0f;               // linspace(0.5, 10, 48)
  const float dv0 = fmaf(lo ? 0.0f : 24.0f, dstep, 0.5f);  // depth of iter 0

  const int w = w16 * 16 + m;
  const long long pixBase = (long long)h * WW + w;
  const long long MASKBASE = (long long)NB * NV * DN * HH * WW * 2;
  const int dbase = lo ? 0 : 24;
  const long long idx0 = (long long)(nv * DN + dbase) * (HH * WW) + pixBase;
  float* __restrict__ outSamp = out + idx0 * 2;
  float* __restrict__ outMask = out + MASKBASE + idx0;

#pragma unroll
  for (int it = 0; it < 24; ++it) {
    const float dv = dv0 + (float)it * dstep;      // (float)it*dstep: literal
    const float px = fmaf(dv, rx, bx);
    const float py = fmaf(dv, ry, by);
    const float pz = fmaf(dv, rz, bz);
    const float zs = (fabsf(pz) < 1e-6f) ? 1e-6f : pz;
    const float inv = 1.0f / zs;                   // IEEE divide sequence
    const float x = px * inv - xs;
    const float y = py * inv - ys;
    const bool msk = (x >= 0.0f) && (x <= (float)(WW - 1)) &&
                     (y >= 0.0f) && (y <= (float)(HH - 1)) && (pz > 1e-6f);
    const float mf = msk ? 1.0f : 0.0f;

    v2f s; s.x = x * mf; s.y = y * mf;
    __builtin_nontemporal_store(s, (v2f*)(outSamp + (long long)it * (HH * WW * 2)));
    __builtin_nontemporal_store(mf, outMask + (long long)it * (HH * WW));
  }
}

extern "C" void kernel_launch(void* const* d_in, const int* in_sizes, int n_in,
                              void* d_out, int out_size, void* d_ws, size_t ws_size,
                              hipStream_t stream) {
  const float* ys_dst = (const float*)d_in[0];
  const float* xs_dst = (const float*)d_in[1];
  const float* ys_src = (const float*)d_in[2];
  const float* xs_src = (const float*)d_in[3];
  // d_in[4] = height, d_in[5] = width (compile-time constants here)
  const float* dstK = (const float*)d_in[6];
  const float* dstE = (const float*)d_in[7];
  const float* srcK = (const float*)d_in[8];
  const float* srcE = (const float*)d_in[9];
  float* out = (float*)d_out;
  float* ws = (float*)d_ws;

  bdvm_setup_kernel<<<1, 32, 0, stream>>>(ys_dst, xs_dst, ys_src, xs_src,
                                          dstK, dstE, srcK, srcE, ws);

  const int strips = NB * NV * HH * (WW / 16);    // 38400 waves
  bdvm_sweep_kernel<<<strips / 8, 256, 0, stream>>>(ws, out);
}